// SalienceWriteHead_2001454760110
// MI455X (gfx1250) — compile-verified
//
#include <hip/hip_runtime.h>
#include <stdint.h>

// SalienceWriteHead for MI455X (gfx1250).
// B=128, T=2048, D=512, H=8, HD=64.
// Single-pass flash-softmax over T: x (512 MB) is read exactly once from HBM
// (~22us at 23.3 TB/s roofline). Logits GEMM runs on v_wmma_f32_16x16x4_f32;
// the x tile is staged in LDS via global_load_async_to_lds_b128 (ASYNCcnt).
// All WMMA-loop operands are zero-padded to 16 columns so the hot loop is
// branch-free (no EXEC save/restore churn around the DS loads).

#define Bb      128
#define Tt      2048
#define Dd      512
#define Hh      8
#define HDd     64
#define CHUNK   128            // t rows per block
#define CHN     (Tt / CHUNK)   // 16 chunks
#define NTHR    256            // 8 waves of 32

typedef __attribute__((ext_vector_type(2))) float v2f;
typedef __attribute__((ext_vector_type(8))) float v8f;

__device__ __forceinline__ void async_copy_b128(uint32_t lds_byte_addr,
                                                uint32_t vaddr_off,
                                                uint64_t sbase) {
  // GVS mode: mem addr = SADDR(64b sgpr) + VADDR(32b vgpr); VDST = LDS byte addr.
  asm volatile("global_load_async_to_lds_b128 %0, %1, %2"
               :: "v"(lds_byte_addr), "v"(vaddr_off), "s"(sbase)
               : "memory");
}

__device__ __forceinline__ void wait_async0() {
  asm volatile("s_wait_asynccnt 0x0" ::: "memory");
}

// ---------------------------------------------------------------------------
// Kernel A: per-(b, chunk) partials: m[h], s[h], v[h*64+d]
// ---------------------------------------------------------------------------
extern "C" __global__ __launch_bounds__(NTHR)
void salience_partial_kernel(const float* __restrict__ x,
                             const int*   __restrict__ mask,
                             const float* __restrict__ temp,
                             const float* __restrict__ w_sal,
                             const float* __restrict__ b_sal,
                             float* __restrict__ pm,
                             float* __restrict__ ps,
                             float* __restrict__ pv) {
  extern __shared__ float smem[];
  float* xs   = smem;                    // CHUNK*Dd = 65536 f (256 KB)
  float* wsal = xs   + CHUNK * Dd;       // Dd*16    = 8192 f (w/temp, cols 8..15 = 0)
  float* lg   = wsal + Dd * 16;          // CHUNK*16 = 2048 f (logits -> exps, stride 16)
  float* bst  = lg   + CHUNK * 16;       // 16 (b_sal/temp, cols 8..15 = 0)
  float* invt = bst  + 16;               // Hh
  float* mx   = invt + Hh;               // Hh (chunk max per head)
  int*   msk  = (int*)(mx + Hh);         // CHUNK

  const int tid  = threadIdx.x;
  const int b    = blockIdx.y;
  const int c    = blockIdx.x;
  const int t0   = c * CHUNK;

  // --- 1) kick off async HBM -> LDS copy of the contiguous 128x512 f32 tile
  const uint64_t gsrc = (uint64_t)(uintptr_t)(x + ((size_t)b * Tt + t0) * Dd);
  const uint32_t lds_base = (uint32_t)(uintptr_t)xs;
  #pragma unroll
  for (int i = 0; i < (CHUNK * Dd * 4) / (NTHR * 16); ++i) {   // 64 iters
    uint32_t boff = (uint32_t)((i * NTHR + tid) * 16);
    async_copy_b128(lds_base + boff, boff, gsrc);
  }

  // --- 2) per-head temperature terms + mask while the DMA runs
  if (tid < Hh) {
    float t  = temp[tid];
    float tp = logf(1.f + expf(t)) + 0.3f;     // softplus + 0.3
    invt[tid] = 1.f / tp;
  }
  if (tid < CHUNK) msk[tid] = mask[(size_t)b * Tt + t0 + tid];
  __syncthreads();                             // invt visible

  // --- 3) zero-padded operands: wsal[k][h<8] = w_sal[k][h]/temp[h], else 0
  for (int i = tid; i < Dd * 16; i += NTHR) {
    int k = i >> 4, h = i & 15;
    wsal[i] = (h < Hh) ? w_sal[k * Hh + h] * invt[h] : 0.f;
  }
  if (tid < 16) bst[tid] = (tid < Hh) ? b_sal[tid] * invt[tid] : 0.f;

  wait_async0();
  __syncthreads();                             // xs + wsal + bst ready

  // --- 4) logits tile per wave via v_wmma_f32_16x16x4_f32 (branch-free loop)
  // A 16x4 f32 layout: lanes 0-15 -> M=lane, K={0,1}; lanes 16-31 -> K={2,3}.
  // B 4x16 f32 layout: lanes 0-15 -> N=lane, K={0,1}; lanes 16-31 -> K={2,3}.
  const int wave  = tid >> 5;
  const int lane  = tid & 31;
  const int trow  = wave * 16;                 // 8 waves cover 128 rows
  const int mrow  = lane & 15;
  const int khalf = lane >> 4;
  const int ncol  = lane & 15;

  v8f acc;
  {
    const float seed = bst[ncol];
    #pragma unroll
    for (int r = 0; r < 8; ++r) acc[r] = seed;
  }

  const float* arow = xs + (trow + mrow) * Dd;
  const float* bcol = wsal + ncol;
  #pragma unroll 4
  for (int k0 = 0; k0 < Dd; k0 += 4) {
    const int kk = k0 + 2 * khalf;
    v2f a;  a.x = arow[kk];           a.y = arow[kk + 1];
    v2f bm; bm.x = bcol[kk * 16];     bm.y = bcol[(kk + 1) * 16];
    acc = __builtin_amdgcn_wmma_f32_16x16x4_f32(false, a, false, bm,
                                                (short)0, acc, false, false);
  }

  // --- 5) masked logits to LDS (D layout: VGPR r, lane l -> M=r+8*khalf, N=l%16)
  //        stride-16 rows; all 32 lanes store (cols 8..15 are never read back)
  #pragma unroll
  for (int r = 0; r < 8; ++r) {
    int t = trow + r + 8 * khalf;
    lg[t * 16 + ncol] = msk[t] ? acc[r] : -1.0e9f;
  }
  __syncthreads();

  // --- 6) chunk max per head (wave h reduces column h)
  {
    float m = -1.0e9f;
    for (int t = lane; t < CHUNK; t += 32) m = fmaxf(m, lg[t * 16 + wave]);
    #pragma unroll
    for (int off = 16; off > 0; off >>= 1) m = fmaxf(m, __shfl_xor(m, off, 32));
    if (lane == 0) {
      mx[wave] = m;
      pm[((size_t)b * CHN + c) * Hh + wave] = m;
    }
  }
  __syncthreads();

  // --- 7) exps (masked) overwrite lg columns 0..7
  for (int i = tid; i < CHUNK * Hh; i += NTHR) {
    int t = i >> 3, h = i & 7;
    lg[t * 16 + h] = msk[t] ? __expf(lg[t * 16 + h] - mx[h]) : 0.f;
  }
  __syncthreads();

  // --- 8) chunk sum per head + weighted-vector accumulation from the SAME
  //         LDS-resident x tile (this is what makes the kernel single-pass)
  {
    float s = 0.f;
    for (int t = lane; t < CHUNK; t += 32) s += lg[t * 16 + wave];
    #pragma unroll
    for (int off = 16; off > 0; off >>= 1) s += __shfl_xor(s, off, 32);
    if (lane == 0) ps[((size_t)b * CHN + c) * Hh + wave] = s;
  }
  {
    const int i0 = tid, i1 = tid + NTHR;       // each thread owns 2 of 512 dims
    const int h0 = i0 >> 6, h1 = i1 >> 6;
    float a0 = 0.f, a1 = 0.f;
    for (int t = 0; t < CHUNK; ++t) {
      a0 = fmaf(lg[t * 16 + h0], xs[t * Dd + i0], a0);
      a1 = fmaf(lg[t * 16 + h1], xs[t * Dd + i1], a1);
    }
    const size_t pidx = (size_t)b * CHN + c;
    pv[pidx * Dd + i0] = a0;
    pv[pidx * Dd + i1] = a1;
  }
}

// ---------------------------------------------------------------------------
// Kernel B: flash-combine across chunks + gate / RMS epilogue
// ---------------------------------------------------------------------------
extern "C" __global__ __launch_bounds__(NTHR)
void salience_finalize_kernel(const float* __restrict__ pm,
                              const float* __restrict__ ps,
                              const float* __restrict__ pv,
                              const float* __restrict__ w_gate,
                              const float* __restrict__ b_gate,
                              const float* __restrict__ scale,
                              float* __restrict__ out) {
  __shared__ float M[Hh], S[Hh], EF[CHN * Hh], WV[Dd], U[Hh], red[8], rinv;
  const int tid = threadIdx.x;
  const int b   = blockIdx.x;

  const float* pmb = pm + (size_t)b * CHN * Hh;
  const float* psb = ps + (size_t)b * CHN * Hh;
  const float* pvb = pv + (size_t)b * CHN * Dd;

  if (tid < Hh) {
    float m = -1.0e9f;
    for (int c = 0; c < CHN; ++c) m = fmaxf(m, pmb[c * Hh + tid]);
    M[tid] = m;
  }
  __syncthreads();
  for (int i = tid; i < CHN * Hh; i += NTHR)
    EF[i] = __expf(pmb[i] - M[i & 7]);
  __syncthreads();
  if (tid < Hh) {
    float s = 0.f;
    for (int c = 0; c < CHN; ++c) s += psb[c * Hh + tid] * EF[c * Hh + tid];
    S[tid] = s;
  }
  const int i0 = tid, i1 = tid + NTHR;
  const int h0 = i0 >> 6, h1 = i1 >> 6;
  float v0 = 0.f, v1 = 0.f;
  for (int c = 0; c < CHN; ++c) {
    v0 = fmaf(pvb[c * Dd + i0], EF[c * Hh + h0], v0);
    v1 = fmaf(pvb[c * Dd + i1], EF[c * Hh + h1], v1);
  }
  __syncthreads();                              // S ready
  const float wv0 = v0 / (S[h0] + 1e-6f);       // weights = exps/(sum+1e-6)
  const float wv1 = v1 / (S[h1] + 1e-6f);
  WV[i0] = wv0;  WV[i1] = wv1;
  __syncthreads();

  const int wave = tid >> 5, lane = tid & 31;
  {   // gate per head: wave h dots WV[h*64..] with w_gate[0..63]
    float g = WV[wave * HDd + lane]      * w_gate[lane]
            + WV[wave * HDd + lane + 32] * w_gate[lane + 32];
    #pragma unroll
    for (int off = 16; off > 0; off >>= 1) g += __shfl_xor(g, off, 32);
    if (lane == 0) {
      float valid = (S[0] > 0.f) ? 1.f : 0.f;   // any(mask[b,:])
      U[wave] = valid / (1.f + __expf(-(g + b_gate[0])));
    }
  }
  {   // RMS over the 512-dim write vector
    float sq = wv0 * wv0 + wv1 * wv1;
    #pragma unroll
    for (int off = 16; off > 0; off >>= 1) sq += __shfl_xor(sq, off, 32);
    if (lane == 0) red[wave] = sq;
  }
  __syncthreads();
  if (tid == 0) {
    float tot = 0.f;
    #pragma unroll
    for (int w = 0; w < 8; ++w) tot += red[w];
    rinv = rsqrtf(tot / (float)Dd + 1e-6f);
  }
  __syncthreads();

  float* out_vec = out;                          // (B, D)
  float* out_ue  = out + (size_t)Bb * Dd;        // (B, D)
  float* out_uh  = out + (size_t)2 * Bb * Dd;    // (B, H)
  out_vec[(size_t)b * Dd + i0] = wv0 * rinv * scale[i0];
  out_vec[(size_t)b * Dd + i1] = wv1 * rinv * scale[i1];
  out_ue [(size_t)b * Dd + i0] = U[h0];
  out_ue [(size_t)b * Dd + i1] = U[h1];
  if (tid < Hh) out_uh[(size_t)b * Hh + tid] = U[tid];
}

// ---------------------------------------------------------------------------
extern "C" void kernel_launch(void* const* d_in, const int* in_sizes, int n_in,
                              void* d_out, int out_size, void* d_ws, size_t ws_size,
                              hipStream_t stream) {
  const float* x      = (const float*)d_in[0];
  const int*   mask   = (const int*)  d_in[1];
  const float* temp   = (const float*)d_in[2];
  const float* w_sal  = (const float*)d_in[3];
  const float* b_sal  = (const float*)d_in[4];
  const float* w_gate = (const float*)d_in[5];
  const float* b_gate = (const float*)d_in[6];
  const float* scale  = (const float*)d_in[7];
  float* out = (float*)d_out;

  // workspace: pm/ps (B*CHN*H each) + pv (B*CHN*D)  ~= 4.3 MB
  float* pm = (float*)d_ws;
  float* ps = pm + (size_t)Bb * CHN * Hh;
  float* pv = ps + (size_t)Bb * CHN * Hh;

  const size_t smem_bytes =
      (size_t)(CHUNK * Dd + Dd * 16 + CHUNK * 16 + 16 + 2 * Hh) * sizeof(float)
      + (size_t)CHUNK * sizeof(int);             // ~304 KB of the 320 KB WGP LDS

  dim3 gridA(CHN, Bb);
  salience_partial_kernel<<<gridA, NTHR, smem_bytes, stream>>>(
      x, mask, temp, w_sal, b_sal, pm, ps, pv);

  salience_finalize_kernel<<<Bb, NTHR, 0, stream>>>(
      pm, ps, pv, w_gate, b_gate, scale, out);
}